// EGAT_Model_27075473834575
// MI455X (gfx1250) — compile-verified
//
#include <hip/hip_runtime.h>
#include <math.h>

typedef __attribute__((ext_vector_type(2))) float v2f;
typedef __attribute__((ext_vector_type(8))) float v8f;

// ---------- ordered-uint encoding for float atomic max ----------
__device__ __forceinline__ unsigned f2ord(float f) {
    unsigned u = __float_as_uint(f);
    return u ^ (unsigned)(((int)u >> 31) | 0x80000000);
}
__device__ __forceinline__ float ord2f(unsigned o) {
    unsigned u = (o & 0x80000000u) ? (o ^ 0x80000000u) : ~o;
    return __uint_as_float(u);
}
#define ORD_NEG_INF 0x007FFFFFu   // f2ord(-inf)

// ---------- fold attention vectors into weight matrices ----------
// Bf1[64][16]: c in 0..2 -> sum_f W1[k][c*32+f]*al1[c][f]; c in 3..5 -> ar1; else 0
// Bf2[96][16]: same with W2/al2/ar2
// Wae[32][16]: c in 0..2 -> We1 folded by ae1; c in 3..5 -> We2 folded by ae2
// WpPad[192][16]: zero-padded Wp (192x10)
__global__ void k_fold(const float* __restrict__ W1, const float* __restrict__ al1,
                       const float* __restrict__ ar1,
                       const float* __restrict__ We1, const float* __restrict__ ae1,
                       const float* __restrict__ W2, const float* __restrict__ al2,
                       const float* __restrict__ ar2,
                       const float* __restrict__ We2, const float* __restrict__ ae2,
                       const float* __restrict__ Wp,
                       float* __restrict__ Bf1, float* __restrict__ Bf2,
                       float* __restrict__ Wae, float* __restrict__ WpPad) {
    int tid = threadIdx.x;
    for (int idx = tid; idx < 64 * 16; idx += blockDim.x) {
        int k = idx >> 4, c = idx & 15;
        float v = 0.f;
        if (c < 3)      { for (int f = 0; f < 32; ++f) v += W1[k * 96 + c * 32 + f] * al1[c * 32 + f]; }
        else if (c < 6) { int h = c - 3; for (int f = 0; f < 32; ++f) v += W1[k * 96 + h * 32 + f] * ar1[h * 32 + f]; }
        Bf1[idx] = v;
    }
    for (int idx = tid; idx < 96 * 16; idx += blockDim.x) {
        int k = idx >> 4, c = idx & 15;
        float v = 0.f;
        if (c < 3)      { for (int f = 0; f < 32; ++f) v += W2[k * 96 + c * 32 + f] * al2[c * 32 + f]; }
        else if (c < 6) { int h = c - 3; for (int f = 0; f < 32; ++f) v += W2[k * 96 + h * 32 + f] * ar2[h * 32 + f]; }
        Bf2[idx] = v;
    }
    for (int idx = tid; idx < 32 * 16; idx += blockDim.x) {
        int k = idx >> 4, c = idx & 15;
        float v = 0.f;
        if (c < 3)      { for (int f = 0; f < 32; ++f) v += We1[k * 96 + c * 32 + f] * ae1[c * 32 + f]; }
        else if (c < 6) { int h = c - 3; for (int f = 0; f < 32; ++f) v += We2[k * 96 + h * 32 + f] * ae2[h * 32 + f]; }
        Wae[idx] = v;
    }
    for (int idx = tid; idx < 192 * 16; idx += blockDim.x) {
        int k = idx >> 4, c = idx & 15;
        WpPad[idx] = (c < 10) ? Wp[k * 10 + c] : 0.f;
    }
}

// ---------- init: rst=0 over n96, m=-inf(ord), s=0 over n3 ----------
__global__ void k_init(unsigned* __restrict__ m, float* __restrict__ s,
                       float* __restrict__ rst, int n3, int n96) {
    int idx = blockIdx.x * blockDim.x + threadIdx.x;
    if (idx < n96) rst[idx] = 0.f;
    if (idx < n3) { m[idx] = ORD_NEG_INF; s[idx] = 0.f; }
}

__global__ void k_reinit(unsigned* __restrict__ m, float* __restrict__ s, int n3) {
    int idx = blockIdx.x * blockDim.x + threadIdx.x;
    if (idx < n3) { m[idx] = ORD_NEG_INF; s[idx] = 0.f; }
}

// ---------- WMMA node projection: ft[N,96]=A@Bw, el/er via folded 7th tile ----------
__global__ void k_proj(const float* __restrict__ A, int K,
                       const float* __restrict__ Bw,   // [K,96] row-major
                       const float* __restrict__ Bf,   // [K,16] folded al|ar
                       float* __restrict__ ft, float* __restrict__ el,
                       float* __restrict__ er, int nrows) {
    int wave = (blockIdx.x * blockDim.x + threadIdx.x) >> 5;
    int lane = threadIdx.x & 31;
    int half = lane >> 4;
    int r    = lane & 15;
    int row0 = wave * 16;
    if (row0 >= nrows) return;               // wave-uniform: EXEC stays all-ones for WMMA

    v8f acc[6];
    v8f accF = {};
    #pragma unroll
    for (int t = 0; t < 6; ++t) acc[t] = (v8f){0, 0, 0, 0, 0, 0, 0, 0};

    const float* Arow = A + (size_t)(row0 + r) * K;  // lanes r and r+16 share row r
    for (int kb = 0; kb < K; kb += 4) {
        int kk = kb + half * 2;
        v2f a; a.x = Arow[kk]; a.y = Arow[kk + 1];
        #pragma unroll
        for (int t = 0; t < 6; ++t) {
            v2f b;
            b.x = Bw[kk * 96 + t * 16 + r];
            b.y = Bw[(kk + 1) * 96 + t * 16 + r];
            acc[t] = __builtin_amdgcn_wmma_f32_16x16x4_f32(false, a, false, b, (short)0,
                                                           acc[t], false, false);
        }
        v2f bf;
        bf.x = Bf[kk * 16 + r];
        bf.y = Bf[(kk + 1) * 16 + r];
        accF = __builtin_amdgcn_wmma_f32_16x16x4_f32(false, a, false, bf, (short)0,
                                                     accF, false, false);
    }
    #pragma unroll
    for (int i = 0; i < 8; ++i) {
        int row = row0 + half * 8 + i;       // D: vgpr i holds rows i / i+8
        #pragma unroll
        for (int t = 0; t < 6; ++t) ft[(size_t)row * 96 + t * 16 + r] = acc[t][i];
        float v = accF[i];
        if (r < 3)       el[row * 3 + r] = v;
        else if (r < 6)  er[row * 3 + (r - 3)] = v;
    }
}

// ---------- WMMA edge projection: ee1/ee2 = ef @ Wae (both layers in one pass) ----------
__global__ void k_ee(const float* __restrict__ ef, const float* __restrict__ Wae,
                     float* __restrict__ ee1, float* __restrict__ ee2, int nE) {
    int wave = (blockIdx.x * blockDim.x + threadIdx.x) >> 5;
    int lane = threadIdx.x & 31;
    int half = lane >> 4;
    int r    = lane & 15;
    int row0 = wave * 16;
    if (row0 >= nE) return;

    v8f acc = {};
    const float* Arow = ef + (size_t)(row0 + r) * 32;
    for (int kb = 0; kb < 32; kb += 4) {
        int kk = kb + half * 2;
        v2f a; a.x = Arow[kk]; a.y = Arow[kk + 1];
        v2f b; b.x = Wae[kk * 16 + r]; b.y = Wae[(kk + 1) * 16 + r];
        acc = __builtin_amdgcn_wmma_f32_16x16x4_f32(false, a, false, b, (short)0,
                                                    acc, false, false);
    }
    #pragma unroll
    for (int i = 0; i < 8; ++i) {
        int row = row0 + half * 8 + i;
        float v = acc[i];
        if (r < 3)      ee1[row * 3 + r] = v;
        else if (r < 6) ee2[row * 3 + (r - 3)] = v;
    }
}

// ---------- attention logits + leaky_relu + segment max (one thread per edge) ----------
__global__ void k_scores(const float* __restrict__ el, const float* __restrict__ er,
                         const float* __restrict__ ee, const int* __restrict__ src,
                         const int* __restrict__ dst, float* __restrict__ esc,
                         unsigned* __restrict__ m, int nE) {
    int e = blockIdx.x * blockDim.x + threadIdx.x;
    if (e >= nE) return;
    int sN = src[e] * 3;
    int dN = dst[e] * 3;
    #pragma unroll
    for (int h = 0; h < 3; ++h) {
        float v = el[sN + h] + er[dN + h] + ee[e * 3 + h];
        v = (v > 0.f) ? v : 0.2f * v;
        esc[e * 3 + h] = v;
        atomicMax(&m[dN + h], f2ord(v));
    }
}

// ---------- exp(score - max) + segment sum (one thread per edge) ----------
__global__ void k_exp(const int* __restrict__ dst, const unsigned* __restrict__ m,
                      float* __restrict__ esc, float* __restrict__ s, int nE) {
    int e = blockIdx.x * blockDim.x + threadIdx.x;
    if (e >= nE) return;
    int dN = dst[e] * 3;
    #pragma unroll
    for (int h = 0; h < 3; ++h) {
        float ex = expf(esc[e * 3 + h] - ord2f(m[dN + h]));
        esc[e * 3 + h] = ex;
        atomicAdd(&s[dN + h], ex);
    }
}

// ---------- weighted aggregation: rst[dst] += ft[src] * a (4 features/thread) ----------
__global__ void k_agg(const int* __restrict__ src, const int* __restrict__ dst,
                      const float* __restrict__ esc, const float* __restrict__ s,
                      const float* __restrict__ ft, float* __restrict__ rst, int total24) {
    int idx = blockIdx.x * blockDim.x + threadIdx.x;
    if (idx >= total24) return;
    int e  = idx / 24;
    int q  = idx - e * 24;
    int f0 = q * 4;                 // 4-feature group; never crosses a 32-wide head
    int h  = f0 >> 5;
    int d  = dst[e];
    float w = esc[e * 3 + h] / s[d * 3 + h];
    const float4 v = *(const float4*)(ft + (size_t)src[e] * 96 + f0);
    float* rp = rst + (size_t)d * 96 + f0;
    atomicAdd(rp + 0, v.x * w);
    atomicAdd(rp + 1, v.y * w);
    atomicAdd(rp + 2, v.z * w);
    atomicAdd(rp + 3, v.w * w);
}

// ---------- h = relu(rst + b1); rst reset for layer 2 (float4) ----------
__global__ void k_fin1(float* __restrict__ rst, const float* __restrict__ b,
                       float* __restrict__ h, int total4) {
    int idx = blockIdx.x * blockDim.x + threadIdx.x;
    if (idx >= total4) return;
    int f0 = (idx * 4) % 96;
    float4 v = *(const float4*)(rst + (size_t)idx * 4);
    const float4 bb = *(const float4*)(b + f0);
    v.x = fmaxf(v.x + bb.x, 0.f);
    v.y = fmaxf(v.y + bb.y, 0.f);
    v.z = fmaxf(v.z + bb.z, 0.f);
    v.w = fmaxf(v.w + bb.w, 0.f);
    *(float4*)(h + (size_t)idx * 4) = v;
    *(float4*)(rst + (size_t)idx * 4) = (float4){0.f, 0.f, 0.f, 0.f};
}

// ---------- h2 = rst + b2 (float4) ----------
__global__ void k_fin2(const float* __restrict__ rst, const float* __restrict__ b,
                       float* __restrict__ h, int total4) {
    int idx = blockIdx.x * blockDim.x + threadIdx.x;
    if (idx >= total4) return;
    int f0 = (idx * 4) % 96;
    float4 v = *(const float4*)(rst + (size_t)idx * 4);
    const float4 bb = *(const float4*)(b + f0);
    v.x += bb.x; v.y += bb.y; v.z += bb.z; v.w += bb.w;
    *(float4*)(h + (size_t)idx * 4) = v;
}

// ---------- WMMA edge score: out = cat(h[src],h[dst]) @ WpPad + bp ----------
__global__ void k_score(const float* __restrict__ h, const int* __restrict__ src,
                        const int* __restrict__ dst, const float* __restrict__ WpPad,
                        const float* __restrict__ bp, float* __restrict__ out, int nE) {
    int wave = (blockIdx.x * blockDim.x + threadIdx.x) >> 5;
    int lane = threadIdx.x & 31;
    int half = lane >> 4;
    int r    = lane & 15;
    int row0 = wave * 16;
    if (row0 >= nE) return;

    const float* hs = h + (size_t)src[row0 + r] * 96;
    const float* hd = h + (size_t)dst[row0 + r] * 96;
    v8f acc = {};
    for (int kb = 0; kb < 192; kb += 4) {
        int kk = kb + half * 2;
        const float* p = (kk < 96) ? (hs + kk) : (hd + (kk - 96));
        v2f a; a.x = p[0]; a.y = p[1];
        v2f b; b.x = WpPad[kk * 16 + r]; b.y = WpPad[(kk + 1) * 16 + r];
        acc = __builtin_amdgcn_wmma_f32_16x16x4_f32(false, a, false, b, (short)0,
                                                    acc, false, false);
    }
    #pragma unroll
    for (int i = 0; i < 8; ++i) {
        int e = row0 + half * 8 + i;
        if (r < 10) out[(size_t)e * 10 + r] = acc[i] + bp[r];
    }
}

extern "C" void kernel_launch(void* const* d_in, const int* in_sizes, int n_in,
                              void* d_out, int out_size, void* d_ws, size_t ws_size,
                              hipStream_t stream) {
    const float* nfeats = (const float*)d_in[0];
    const float* efeats = (const float*)d_in[1];
    const int*   src    = (const int*)d_in[2];
    const int*   dst    = (const int*)d_in[3];
    const float* W1  = (const float*)d_in[4];
    const float* We1 = (const float*)d_in[5];
    const float* al1 = (const float*)d_in[6];
    const float* ar1 = (const float*)d_in[7];
    const float* ae1 = (const float*)d_in[8];
    const float* b1  = (const float*)d_in[9];
    const float* W2  = (const float*)d_in[10];
    const float* We2 = (const float*)d_in[11];
    const float* al2 = (const float*)d_in[12];
    const float* ar2 = (const float*)d_in[13];
    const float* ae2 = (const float*)d_in[14];
    const float* b2  = (const float*)d_in[15];
    const float* Wp  = (const float*)d_in[16];
    const float* bp  = (const float*)d_in[17];

    const int N = in_sizes[0] / 64;
    const int E = in_sizes[2];

    // bump allocator over workspace
    char* w = (char*)d_ws;
    auto alloc = [&](size_t bytes) -> void* {
        void* p = (void*)w;
        w += (bytes + 255) & ~(size_t)255;
        return p;
    };
    float*    ft    = (float*)alloc((size_t)N * 96 * 4);
    float*    hb    = (float*)alloc((size_t)N * 96 * 4);
    float*    rst   = (float*)alloc((size_t)N * 96 * 4);
    float*    el    = (float*)alloc((size_t)N * 3 * 4);
    float*    er    = (float*)alloc((size_t)N * 3 * 4);
    unsigned* mbuf  = (unsigned*)alloc((size_t)N * 3 * 4);
    float*    sbuf  = (float*)alloc((size_t)N * 3 * 4);
    float*    ee1   = (float*)alloc((size_t)E * 3 * 4);
    float*    ee2   = (float*)alloc((size_t)E * 3 * 4);
    float*    esc   = (float*)alloc((size_t)E * 3 * 4);
    float*    Bf1   = (float*)alloc(64 * 16 * 4);
    float*    Bf2   = (float*)alloc(96 * 16 * 4);
    float*    Wae   = (float*)alloc(32 * 16 * 4);
    float*    WpPad = (float*)alloc(192 * 16 * 4);

    float* outp = (float*)d_out;

    const int T = 256;
    const int tilesN = (N + 15) / 16;
    const int tilesE = (E + 15) / 16;
    const int gN96 = (N * 96 + T - 1) / T;
    const int gN4  = (N * 24 + T - 1) / T;     // N*96/4 vector threads
    const int gN3  = (N * 3 + T - 1) / T;
    const int gE   = (E + T - 1) / T;
    const int gE24 = (int)(((size_t)E * 24 + T - 1) / T);

    k_fold<<<1, 256, 0, stream>>>(W1, al1, ar1, We1, ae1, W2, al2, ar2, We2, ae2, Wp,
                                  Bf1, Bf2, Wae, WpPad);
    k_init<<<gN96, T, 0, stream>>>(mbuf, sbuf, rst, N * 3, N * 96);

    // ---- layer 1 ----
    k_proj<<<(tilesN + 7) / 8, T, 0, stream>>>(nfeats, 64, W1, Bf1, ft, el, er, N);
    k_ee<<<(tilesE + 7) / 8, T, 0, stream>>>(efeats, Wae, ee1, ee2, E);
    k_scores<<<gE, T, 0, stream>>>(el, er, ee1, src, dst, esc, mbuf, E);
    k_exp<<<gE, T, 0, stream>>>(dst, mbuf, esc, sbuf, E);
    k_agg<<<gE24, T, 0, stream>>>(src, dst, esc, sbuf, ft, rst, E * 24);
    k_fin1<<<gN4, T, 0, stream>>>(rst, b1, hb, N * 24);    // h=relu(rst+b1); rst=0
    k_reinit<<<gN3, T, 0, stream>>>(mbuf, sbuf, N * 3);

    // ---- layer 2 ----
    k_proj<<<(tilesN + 7) / 8, T, 0, stream>>>(hb, 96, W2, Bf2, ft, el, er, N);
    k_scores<<<gE, T, 0, stream>>>(el, er, ee2, src, dst, esc, mbuf, E);
    k_exp<<<gE, T, 0, stream>>>(dst, mbuf, esc, sbuf, E);
    k_agg<<<gE24, T, 0, stream>>>(src, dst, esc, sbuf, ft, rst, E * 24);
    k_fin2<<<gN4, T, 0, stream>>>(rst, b2, hb, N * 24);    // h2 = rst + b2

    // ---- per-edge MLP score via WMMA ----
    k_score<<<(tilesE + 7) / 8, T, 0, stream>>>(hb, src, dst, WpPad, bp, outp, E);
}